// ProjGA_57724360458831
// MI455X (gfx1250) — compile-verified
//
#include <hip/hip_runtime.h>
#include <hip/hip_bf16.h>

// ============================================================================
// ProjGA on MI455X (gfx1250, wave32, WMMA + Tensor Data Mover).
//
// Shapes: B=8 channels, BATCH=4096 problems, N=16, M=128, L1=16, NITER=5.
// All matmuls are (16 x K) @ (K x 16), K in {16,128} -> chained
// v_wmma_f32_16x16x4_f32 (f32 data, well-conditioned SPD inverse -> keep f32).
//
// Decomposition: 1 block (8 wave32) per batch problem; wave b owns channel b.
// h (8 x 16x128 = 64KB/block) is DMA'd into LDS ONCE by the TDM
// (tensor_load_to_lds) with hardware row padding (128 DW + 4 DW pad ->
// 132-float pitch => conflict-free WMMA fragment loads), then all operands
// stream from LDS. ~126KB LDS/block -> 2 blocks/WGP.
//
// Fragment layouts (ISA 7.12.2, wave32):
//   C/D (16x16 f32, 8 VGPR): elem(M = r + 8*hi, N = lane%16) in c[r]
//   A   (16x4  f32, 2 VGPR): elem(M = lane%16, K = kk + 2*hi + {0,1})
//   B   (4x16  f32, 2 VGPR): elem(K = kk + 2*hi + {0,1}, N = lane%16)
// ============================================================================

typedef __attribute__((ext_vector_type(2))) float v2f;
typedef __attribute__((ext_vector_type(8))) float v8f;
typedef __attribute__((ext_vector_type(4))) unsigned int v4u;
typedef __attribute__((ext_vector_type(8))) int v8i;
typedef __attribute__((ext_vector_type(4))) int v4i;

constexpr int kB = 8, kBatch = 4096, kN = 16, kM = 128, kL = 16, kIter = 5;
constexpr int kHP = 132;  // LDS pitch of h rows (128 + 4 pad DWORDs from TDM)

__device__ __forceinline__ void lds_fence() {
  asm volatile("s_wait_dscnt 0" ::: "memory");
}

__device__ __forceinline__ float wave_sum(float v) {
#pragma unroll
  for (int off = 16; off > 0; off >>= 1) v += __shfl_xor(v, off, 32);
  return v;
}

template <bool TA>
__device__ __forceinline__ v2f load_a_frag(const float* A, int lda, int kk, int q, int hi) {
  const int k0 = kk + 2 * hi;
  v2f a;
  if (!TA) { a.x = A[q * lda + k0];  a.y = A[q * lda + k0 + 1]; }
  else     { a.x = A[k0 * lda + q];  a.y = A[(k0 + 1) * lda + q]; }
  return a;
}

template <bool TB>
__device__ __forceinline__ v2f load_b_frag(const float* B, int ldb, int kk, int q, int hi) {
  const int k0 = kk + 2 * hi;
  v2f b;
  if (!TB) { b.x = B[k0 * ldb + q];  b.y = B[(k0 + 1) * ldb + q]; }
  else     { b.x = B[q * ldb + k0];  b.y = B[q * ldb + k0 + 1]; }
  return b;
}

// C += A(16xK) @ B(Kx16); TA/TB: operand stored transposed in memory.
template <bool TA, bool TB>
__device__ __forceinline__ v8f mma16(const float* A, int lda, const float* B, int ldb, int K,
                                     v8f c, int q, int hi) {
  for (int kk = 0; kk < K; kk += 4) {
    v2f a = load_a_frag<TA>(A, lda, kk, q, hi);
    v2f b = load_b_frag<TB>(B, ldb, kk, q, hi);
    c = __builtin_amdgcn_wmma_f32_16x16x4_f32(false, a, false, b, (short)0, c, false, false);
  }
  return c;
}

__device__ __forceinline__ void store_c(float* D, int ldd, v8f c, int q, int hi) {
#pragma unroll
  for (int r = 0; r < 8; ++r) D[(r + 8 * hi) * ldd + q] = c[r];
}

__device__ __forceinline__ v8f add_identity(v8f c, int q, int hi) {
#pragma unroll
  for (int r = 0; r < 8; ++r) c[r] += ((r + 8 * hi) == q) ? 1.0f : 0.0f;
  return c;
}

// In-wave Gauss-Jordan of a 16x16 SPD matrix in aug[16][33] (cols 16..31 get
// the inverse). Lane L owns column L. Returns logdet (= sum log pivots).
__device__ float invert16(float* aug, int lane) {
  if (lane < 16) {
#pragma unroll
    for (int c = 0; c < 16; ++c) aug[lane * 33 + 16 + c] = (lane == c) ? 1.0f : 0.0f;
  }
  lds_fence();
  float logdet = 0.0f;
  for (int j = 0; j < 16; ++j) {
    float piv = aug[j * 33 + j];
    logdet += __logf(fabsf(piv));
    float rj = aug[j * 33 + lane] / piv;   // scaled pivot row, this lane's column
    aug[j * 33 + lane] = rj;
    lds_fence();
#pragma unroll 4
    for (int rr = 0; rr < 16; ++rr) {
      if (rr == j) continue;
      float f = aug[rr * 33 + j];          // read before lane j zeroes it
      aug[rr * 33 + lane] -= f * rj;
    }
    lds_fence();
  }
  return logdet;
}

// ||wa(128x16) @ wd(16x16)||_F^2, one wave does all 8 row-tiles.
__device__ __forceinline__ float fro_wa_wd(const float* wa, const float* wd, int q, int hi) {
  float s = 0.0f;
#pragma unroll
  for (int mt = 0; mt < 8; ++mt) {
    v8f c = {};
    c = mma16<false, false>(wa + mt * 16 * kL, kL, wd, kN, kL, c, q, hi);
#pragma unroll
    for (int r = 0; r < 8; ++r) s += c[r] * c[r];
  }
  return wave_sum(s);
}

__global__ __launch_bounds__(256)
void projga_kernel(const float* __restrict__ hg, const float* __restrict__ wd0g,
                   const float* __restrict__ hypg, float* __restrict__ out_obj,
                   float* __restrict__ out_wa, float* __restrict__ out_wd) {
  __shared__ float s_h[kB][kN][kHP];        // TDM destination, 132-float pitch
  __shared__ float s_wa[kM][kL];            // shared across channels (b-invariant)
  __shared__ float s_wd[kB][kN][kN];
  __shared__ float s_abar[kN][kM];          // init-only: mean_b(h)
  __shared__ float s_p3[kB][kN][kN];        // per-wave 16x16 staging
  __shared__ float s_red[kB][kN][kN];       // per-wave 16x16 staging / K-partials
  __shared__ float s_aug[kB][16 * 33];      // Gauss-Jordan panels
  __shared__ float s_p[kN][kN];             // p = h_i @ wa (shared)
  __shared__ float s_fro[kB];
  __shared__ float s_scale, s_obj;

  const int bi = blockIdx.x;
  const int tid = threadIdx.x;
  const int w = tid >> 5;        // wave id == channel id b
  const int lane = tid & 31;
  const int q = lane & 15;
  const int hi = lane >> 4;
  const size_t hstr = (size_t)kN * kM;  // 2048 floats per (b,batch) slice of h

  // ====================== stage h into LDS via the TDM ======================
  // 2-D D# (ISA 8.3-8.6): tile = 8 "rows" (channels) x 2048 f32, row stride
  // BATCH*2048 elements; pad_enable inserts 4 DWORDs after every 128 DWORDs,
  // so each 128-float h row lands on a 132-float LDS pitch (bank-conflict-free
  // WMMA fragment loads: pitch%64 words == 4).
#if __has_builtin(__builtin_amdgcn_tensor_load_to_lds)
  if (w == 0) {
    const unsigned long long ga =
        (unsigned long long)(const void*)(hg + (size_t)bi * hstr);
    const unsigned lds_addr = (unsigned)(unsigned long long)(const void*)&s_h[0][0][0];

    v4u g0;
    g0.x = 1u;                                    // count=1, user descriptor
    g0.y = lds_addr;                              // lds_addr [63:32]
    g0.z = (unsigned)(ga & 0xffffffffull);        // global_addr [95:64]
    g0.w = (unsigned)((ga >> 32) & 0x01ffffffull) // global_addr [120:96]
           | (2u << 30);                          // type=2 ("image")

    v8i g1;
    g1[0] = (int)((2u << 16)        // data_size=4B
                | (1u << 20)        // pad_enable
                | (6u << 22)        // pad_interval: 128 DWORDs
                | (3u << 25));      // pad_amount: 4 DWORDs
    g1[1] = (int)(2048u << 16);     // tensor_dim0[15:0] = 2048
    g1[2] = (int)(8u << 16);        // tensor_dim0 hi=0 | tensor_dim1[15:0]=8
    g1[3] = (int)(2048u << 16);     // tensor_dim1 hi=0 | tile_dim0=2048
    g1[4] = 8;                      // tile_dim1=8, tile_dim2=0
    g1[5] = (int)(kBatch * 2048u);  // tensor_dim0_stride lo = 8388608
    g1[6] = 0;                      // stride hi=0 | tensor_dim1_stride lo=0
    g1[7] = 0;

    v4i g2 = {0, 0, 0, 0};          // 2-D tile: dims 2/3 unused
    v4i g3 = {0, 0, 0, 0};
    v8i g4 = {0, 0, 0, 0, 0, 0, 0, 0};  // 6-arg toolchain: extra group, unused

    __builtin_amdgcn_tensor_load_to_lds(g0, g1, g2, g3, g4, 0);
  }
#else
  for (int idx = tid; idx < kB * kN * kM; idx += 256) {
    int b = idx >> 11, rem = idx & 2047, r = rem >> 7, c = rem & 127;
    s_h[b][r][c] = hg[((size_t)b * kBatch + bi) * hstr + rem];
  }
#endif

  // ---- load wd_init (overlaps with the TDM transfer) ----
  for (int idx = tid; idx < kB * kN * kN; idx += 256) {
    int b = idx >> 8, e = idx & 255;
    (&s_wd[0][0][0])[idx] = wd0g[((size_t)b * kBatch + bi) * 256 + e];
  }

#if __has_builtin(__builtin_amdgcn_tensor_load_to_lds)
  if (w == 0) __builtin_amdgcn_s_wait_tensorcnt(0);
#endif
  __syncthreads();   // publish h (and wd) to all waves

  // ---- abar = mean_b h[b]  (16x128), from LDS ----
  for (int idx = tid; idx < kN * kM; idx += 256) {
    int r = idx >> 7, c = idx & 127;
    float s = 0.0f;
#pragma unroll
    for (int b = 0; b < kB; ++b) s += s_h[b][r][c];
    s_abar[r][c] = s * (1.0f / kB);
  }
  __syncthreads();

  // ---- wa init: orthonormal row-space basis of abar via MGS (the reference
  //      SVD's null-space completion is LAPACK-implementation-defined). ----
  if (w == 0) {
    float* ab = &s_abar[0][0];
    for (int k = 0; k < 16; ++k) {
      for (int j = 0; j < k; ++j) {
        float d = 0.0f;
        for (int c = lane; c < kM; c += 32) d += ab[k * kM + c] * ab[j * kM + c];
        d = wave_sum(d);
        for (int c = lane; c < kM; c += 32) ab[k * kM + c] -= d * ab[j * kM + c];
        lds_fence();
      }
      float nn = 0.0f;
      for (int c = lane; c < kM; c += 32) { float v = ab[k * kM + c]; nn += v * v; }
      nn = wave_sum(nn);
      float inv = rsqrtf(fmaxf(nn, 1e-30f));
      for (int c = lane; c < kM; c += 32) ab[k * kM + c] *= inv;
      lds_fence();
    }
    for (int idx = lane; idx < kN * kM; idx += 32) {
      int k = idx >> 7, m = idx & 127;
      s_wa[m][k] = ab[idx];                  // wa column k = k-th basis vector
    }
  }
  __syncthreads();

  // ---- initial wd renorm: wd *= sqrt(N*B / sum_b ||wa@wd_b||^2) ----
  {
    float fr = fro_wa_wd(&s_wa[0][0], &s_wd[w][0][0], q, hi);
    if (lane == 0) s_fro[w] = fr;
  }
  __syncthreads();
  if (tid == 0) {
    float tot = 0.0f;
    for (int j = 0; j < kB; ++j) tot += s_fro[j];
    s_scale = sqrtf((float)(kN * kB) / tot);
  }
  __syncthreads();
  {
    float sc = s_scale;
    for (int idx = tid; idx < kB * 256; idx += 256) (&s_wd[0][0][0])[idx] *= sc;
    if (tid < kB) s_fro[tid] *= sc * sc;
  }
  __syncthreads();

  for (int x = 0; x < kIter; ++x) {
    // ================= wa gradient step =================
    {
      const float* hb = &s_h[w][0][0];
      v8f hw = {};                                   // hw = h_b @ wa   (K=128)
      hw = mma16<false, false>(hb, kHP, &s_wa[0][0], kL, kM, hw, q, hi);
      store_c(&s_red[w][0][0], kN, hw, q, hi); lds_fence();
      v8f t = {};                                    // t = hw @ wd_b
      t = mma16<false, false>(&s_red[w][0][0], kN, &s_wd[w][0][0], kN, kN, t, q, hi);
      store_c(&s_p3[w][0][0], kN, t, q, hi); lds_fence();
      v8f cv = {};                                   // cov = I + t t^T
      cv = mma16<false, true>(&s_p3[w][0][0], kN, &s_p3[w][0][0], kN, kN, cv, q, hi);
      cv = add_identity(cv, q, hi);
      store_c(&s_aug[w][0], 33, cv, q, hi); lds_fence();
      invert16(&s_aug[w][0], lane);
      const float* inv = &s_aug[w][16];              // 16x16, ld=33
      v8f p1 = {};                                   // p1 = inv^T @ hw
      p1 = mma16<true, false>(inv, 33, &s_red[w][0][0], kN, kN, p1, q, hi);
      store_c(&s_p3[w][0][0], kN, p1, q, hi); lds_fence();
      v8f p2 = {};                                   // p2 = p1 @ wd_b
      p2 = mma16<false, false>(&s_p3[w][0][0], kN, &s_wd[w][0][0], kN, kN, p2, q, hi);
      store_c(&s_p3[w][0][0], kN, p2, q, hi); lds_fence();
      v8f p3v = {};                                  // p3 = p2 @ wd_b^T
      p3v = mma16<false, true>(&s_p3[w][0][0], kN, &s_wd[w][0][0], kN, kN, p3v, q, hi);
      store_c(&s_p3[w][0][0], kN, p3v, q, hi);
    }
    __syncthreads();
    {
      // f2 row-tile w:  sum_b  h_b^T[16w..16w+15, :] @ p3_b    (K=16 each)
      v8f f2 = {};
      for (int b = 0; b < kB; ++b) {
        f2 = mma16<true, false>(&s_h[b][0][16 * w], kHP, &s_p3[b][0][0], kN, kN, f2, q, hi);
      }
      const float step0 = hypg[x * (kB + 1)] * (1.0f / kB);
#pragma unroll
      for (int r = 0; r < 8; ++r) s_wa[16 * w + r + 8 * hi][q] += step0 * f2[r];
    }
    __syncthreads();
    // renorm wa
    {
      float fr = fro_wa_wd(&s_wa[0][0], &s_wd[w][0][0], q, hi);
      if (lane == 0) s_fro[w] = fr;
    }
    __syncthreads();
    if (tid == 0) {
      float tot = 0.0f;
      for (int j = 0; j < kB; ++j) tot += s_fro[j];
      s_scale = sqrtf((float)(kN * kB) / tot);
    }
    __syncthreads();
    {
      float sc = s_scale;
      for (int idx = tid; idx < kM * kL; idx += 256) (&s_wa[0][0])[idx] *= sc;
      if (tid < kB) s_fro[tid] *= sc * sc;     // ||wa@wd|| scales with wa
    }
    __syncthreads();

    // ============ sequential projected wd updates (i = 0..B-1) ============
    for (int i = 0; i < kB; ++i) {
      // p = h_i @ wa, cooperatively: wave w does K-chunk [16w, 16w+16)
      {
        v8f pp = {};
        pp = mma16<false, false>(&s_h[i][0][16 * w], kHP, &s_wa[16 * w][0], kL, 16, pp, q, hi);
        store_c(&s_red[w][0][0], kN, pp, q, hi);
      }
      __syncthreads();
      {
        int r = tid >> 4, cc = tid & 15;       // 256 threads, 256 elements
        float acc = 0.0f;
#pragma unroll
        for (int ww = 0; ww < kB; ++ww) acc += s_red[ww][r][cc];
        s_p[r][cc] = acc;
      }
      __syncthreads();
      if (w == 0) {
        v8f t = {};                                  // t = p @ wd_i
        t = mma16<false, false>(&s_p[0][0], kN, &s_wd[i][0][0], kN, kN, t, q, hi);
        store_c(&s_p3[0][0][0], kN, t, q, hi); lds_fence();
        v8f cv = {};                                 // cov = I + t t^T
        cv = mma16<false, true>(&s_p3[0][0][0], kN, &s_p3[0][0][0], kN, kN, cv, q, hi);
        cv = add_identity(cv, q, hi);
        store_c(&s_aug[0][0], 33, cv, q, hi); lds_fence();
        invert16(&s_aug[0][0], lane);
        const float* inv = &s_aug[0][16];
        v8f g1 = {};                                 // g1 = p^T @ inv^T
        g1 = mma16<true, true>(&s_p[0][0], kN, inv, 33, kN, g1, q, hi);
        store_c(&s_p3[0][0][0], kN, g1, q, hi); lds_fence();
        v8f g2 = {};                                 // g2 = g1 @ p
        g2 = mma16<false, false>(&s_p3[0][0][0], kN, &s_p[0][0], kN, kN, g2, q, hi);
        store_c(&s_p3[0][0][0], kN, g2, q, hi); lds_fence();
        v8f g = {};                                  // g = g2 @ wd_i
        g = mma16<false, false>(&s_p3[0][0][0], kN, &s_wd[i][0][0], kN, kN, g, q, hi);
        const float stepi = hypg[x * (kB + 1) + i + 1] * (1.0f / kB);
#pragma unroll
        for (int r = 0; r < 8; ++r) s_wd[i][r + 8 * hi][q] += stepi * g[r];
        if (lane == 0) s_fro[i] = 0.0f;
      }
      __syncthreads();
      {
        // fresh fro2[i] = ||wa @ wd_i||^2, wave w does row-tile w
        v8f c = {};
        c = mma16<false, false>(&s_wa[16 * w][0], kL, &s_wd[i][0][0], kN, 16, c, q, hi);
        float s = 0.0f;
#pragma unroll
        for (int r = 0; r < 8; ++r) s += c[r] * c[r];
        s = wave_sum(s);
        if (lane == 0) atomicAdd(&s_fro[i], s);
      }
      __syncthreads();
      if (tid == 0) {
        float tot = 0.0f;
        for (int j = 0; j < kB; ++j) tot += s_fro[j];
        s_scale = sqrtf((float)(kN * kB) / tot);
      }
      __syncthreads();
      {
        float sc = s_scale;
        for (int idx = tid; idx < kB * 256; idx += 256) (&s_wd[0][0][0])[idx] *= sc;
        if (tid < kB) s_fro[tid] *= sc * sc;
      }
      __syncthreads();
    }

    // ================= objective: sum_b logdet(cov_b) / B =================
    if (tid == 0) s_obj = 0.0f;
    __syncthreads();
    {
      const float* hb = &s_h[w][0][0];
      v8f p = {};
      p = mma16<false, false>(hb, kHP, &s_wa[0][0], kL, kM, p, q, hi);
      store_c(&s_red[w][0][0], kN, p, q, hi); lds_fence();
      v8f t = {};
      t = mma16<false, false>(&s_red[w][0][0], kN, &s_wd[w][0][0], kN, kN, t, q, hi);
      store_c(&s_p3[w][0][0], kN, t, q, hi); lds_fence();
      v8f cv = {};
      cv = mma16<false, true>(&s_p3[w][0][0], kN, &s_p3[w][0][0], kN, kN, cv, q, hi);
      cv = add_identity(cv, q, hi);
      store_c(&s_aug[w][0], 33, cv, q, hi); lds_fence();
      float ld = invert16(&s_aug[w][0], lane);
      if (lane == 0) atomicAdd(&s_obj, ld);
    }
    __syncthreads();
    if (tid == 0) out_obj[(size_t)bi * kIter + x] = s_obj * (1.0f / kB);
    __syncthreads();
  }

  // ---- outputs: obj written above; wa broadcast over b; wd per channel ----
  for (int idx = tid; idx < kB * kM * kL; idx += 256) {
    int b = idx >> 11, rem = idx & 2047;
    out_wa[((size_t)b * kBatch + bi) * (kM * kL) + rem] = (&s_wa[0][0])[rem];
  }
  for (int idx = tid; idx < kB * kN * kN; idx += 256) {
    int b = idx >> 8, e = idx & 255;
    out_wd[((size_t)b * kBatch + bi) * 256 + e] = (&s_wd[0][0][0])[idx];
  }
}

extern "C" void kernel_launch(void* const* d_in, const int* in_sizes, int n_in,
                              void* d_out, int out_size, void* d_ws, size_t ws_size,
                              hipStream_t stream) {
  (void)in_sizes; (void)n_in; (void)out_size; (void)d_ws; (void)ws_size;
  const float* h   = (const float*)d_in[0];
  const float* wd0 = (const float*)d_in[1];
  const float* hyp = (const float*)d_in[2];
  float* obj = (float*)d_out;
  float* wa  = obj + (size_t)kBatch * kIter;
  float* wd  = wa + (size_t)kB * kBatch * kM * kL;
  projga_kernel<<<dim3(kBatch), dim3(256), 0, stream>>>(h, wd0, hyp, obj, wa, wd);
}